// TransNonlinear_LSTM_11089605558940
// MI455X (gfx1250) — compile-verified
//
#include <hip/hip_runtime.h>

typedef float v2f __attribute__((ext_vector_type(2)));
typedef float v8f __attribute__((ext_vector_type(8)));

#define BSZ   2048
#define TLEN  10
#define KBLK  150     // independent LSTM blocks
#define HDIM  12      // hidden per block
#define GDIM  48      // 4*HDIM gates (i,f,g,o)
#define NN    1800    // KBLK*HDIM
#define WAVES_PER_WG 8

__device__ __forceinline__ float sigmoid_f(float x) {
    float e = __builtin_amdgcn_exp2f(-1.4426950408889634f * x);
    return __builtin_amdgcn_rcpf(1.0f + e);
}
__device__ __forceinline__ float tanh_f(float x) {
    // tanh(x) = 2*sigmoid(2x) - 1
    float e = __builtin_amdgcn_exp2f(-2.8853900817779268f * x);
    return 2.0f * __builtin_amdgcn_rcpf(1.0f + e) - 1.0f;
}

__global__ __launch_bounds__(WAVES_PER_WG * 32)
void lstm_block_wmma_kernel(const float* __restrict__ v,
                            const float* __restrict__ dx,
                            const float* __restrict__ W_ih,
                            const float* __restrict__ W_hh,
                            const float* __restrict__ b_ih,
                            const float* __restrict__ b_hh,
                            float* __restrict__ out)
{
    // Per-wave private LDS slices (no cross-wave sharing -> no __syncthreads)
    __shared__ float lds_gates[WAVES_PER_WG][16 * GDIM]; // 16 batch rows x 48 gates
    __shared__ float lds_h[WAVES_PER_WG][16 * HDIM];     // 16 batch rows x 12 hidden

    const int wslot = threadIdx.x >> 5;
    const int lane  = threadIdx.x & 31;
    const int wave  = blockIdx.x * WAVES_PER_WG + wslot; // 0 .. 19199
    const int k     = wave % KBLK;   // LSTM block id
    const int bt    = wave / KBLK;   // batch tile id (16 rows each)
    const int p     = lane & 15;     // position within half-wave
    const int lh    = lane >> 4;     // half-wave id

    float* gbuf = lds_gates[wslot];
    float* hbuf = lds_h[wslot];

    // ---- Preload W_hh as 3x3 B-fragments (4x16 f32 each) ----
    // B layout: vgpr j, half lh -> K = kt*4 + lh*2 + j ; N = p (gate column)
    v2f bf[3][3];
    #pragma unroll
    for (int nt = 0; nt < 3; ++nt) {
        const int g = nt * 16 + p;
        #pragma unroll
        for (int kt = 0; kt < 3; ++kt) {
            const int h0i = kt * 4 + lh * 2;
            const float* wp = W_hh + ((size_t)k * GDIM + g) * HDIM + h0i;
            bf[nt][kt].x = wp[0];
            bf[nt][kt].y = wp[1];
        }
    }

    // ---- Preload input projection weights + fused bias for my gate column ----
    float wi0[3], wi1[3], bias[3];
    #pragma unroll
    for (int nt = 0; nt < 3; ++nt) {
        const int g = nt * 16 + p;
        wi0[nt]  = W_ih[((size_t)k * GDIM + g) * 2 + 0];
        wi1[nt]  = W_ih[((size_t)k * GDIM + g) * 2 + 1];
        bias[nt] = b_ih[(size_t)k * GDIM + g] + b_hh[(size_t)k * GDIM + g];
    }

    // ---- Initial h (A fragments) from v: h0[b,k,h] = v[b*1800 + h*150 + k] ----
    // A layout: vgpr {x,y}, half lh -> M = p ; K = kt*4 + lh*2 + {0,1}
    v2f a[3];
    #pragma unroll
    for (int kt = 0; kt < 3; ++kt) {
        const int h0i  = kt * 4 + lh * 2;
        const size_t brow = (size_t)(bt * 16 + p);
        a[kt].x = v[brow * NN + (size_t)(h0i + 0) * KBLK + k];
        a[kt].y = v[brow * NN + (size_t)(h0i + 1) * KBLK + k];
    }

    // c-state: 6 elements per lane; row = lane>>1, h = 6*(lane&1)+q
    float cst[6];
    #pragma unroll
    for (int q = 0; q < 6; ++q) cst[q] = 0.0f;

    const int erow = lane >> 1;          // batch row (0..15) this lane updates
    const int eh0  = (lane & 1) * 6;     // first hidden index this lane updates

    for (int t = 0; t < TLEN; ++t) {
        // ---- C init = x_proj (dx @ W_ih + bias) in C-fragment layout ----
        // C layout: vgpr j, half lh -> M = lh*8 + j ; N = p
        float d0[8], d1[8];
        #pragma unroll
        for (int j = 0; j < 8; ++j) {
            const size_t brow = (size_t)(bt * 16 + lh * 8 + j);
            const float* dp = dx + (brow * TLEN + t) * 2;
            d0[j] = dp[0];
            d1[j] = dp[1];
        }
        v8f acc[3];
        #pragma unroll
        for (int nt = 0; nt < 3; ++nt) {
            #pragma unroll
            for (int j = 0; j < 8; ++j)
                acc[nt][j] = d0[j] * wi0[nt] + d1[j] * wi1[nt] + bias[nt];
        }

        // ---- 9x V_WMMA_F32_16X16X4_F32: gates += h @ W_hh^T ----
        #pragma unroll
        for (int nt = 0; nt < 3; ++nt) {
            #pragma unroll
            for (int kt = 0; kt < 3; ++kt) {
                acc[nt] = __builtin_amdgcn_wmma_f32_16x16x4_f32(
                    false, a[kt], false, bf[nt][kt],
                    (short)0, acc[nt], false, false);
            }
        }

        // ---- Scatter gates to LDS (C layout -> row-major 16x48) ----
        #pragma unroll
        for (int nt = 0; nt < 3; ++nt) {
            #pragma unroll
            for (int j = 0; j < 8; ++j)
                gbuf[(lh * 8 + j) * GDIM + nt * 16 + p] = acc[nt][j];
        }
        asm volatile("s_wait_dscnt 0" ::: "memory"); // wave-internal LDS ordering

        // ---- Pointwise LSTM cell: 6 (row,h) elements per lane ----
        #pragma unroll
        for (int q = 0; q < 6; ++q) {
            const int hh = eh0 + q;               // 0..11
            const float* gp = gbuf + erow * GDIM + hh;
            const float ig = sigmoid_f(gp[0]);
            const float fg = sigmoid_f(gp[12]);
            const float gg = tanh_f(gp[24]);
            const float og = sigmoid_f(gp[36]);
            const float cv = fg * cst[q] + ig * gg;
            cst[q] = cv;
            const float hv = og * tanh_f(cv);
            hbuf[erow * HDIM + hh] = hv;
            // out[b, t, k*12 + h]
            const size_t brow = (size_t)(bt * 16 + erow);
            out[brow * (TLEN * NN) + (size_t)t * NN + k * HDIM + hh] = hv;
        }
        asm volatile("s_wait_dscnt 0" ::: "memory");

        // ---- Rebuild A fragments for next timestep from LDS h ----
        #pragma unroll
        for (int kt = 0; kt < 3; ++kt) {
            const int h0i = kt * 4 + lh * 2;
            a[kt].x = hbuf[p * HDIM + h0i + 0];
            a[kt].y = hbuf[p * HDIM + h0i + 1];
        }
    }
}

extern "C" void kernel_launch(void* const* d_in, const int* in_sizes, int n_in,
                              void* d_out, int out_size, void* d_ws, size_t ws_size,
                              hipStream_t stream) {
    (void)in_sizes; (void)n_in; (void)out_size; (void)d_ws; (void)ws_size;
    const float* v    = (const float*)d_in[0];
    const float* dx   = (const float*)d_in[1];
    const float* W_ih = (const float*)d_in[2];
    const float* W_hh = (const float*)d_in[3];
    const float* b_ih = (const float*)d_in[4];
    const float* b_hh = (const float*)d_in[5];
    float* out = (float*)d_out;

    const int num_waves = (BSZ / 16) * KBLK;       // 128 * 150 = 19200
    const int blocks = num_waves / WAVES_PER_WG;   // 2400
    lstm_block_wmma_kernel<<<blocks, WAVES_PER_WG * 32, 0, stream>>>(
        v, dx, W_ih, W_hh, b_ih, b_hh, out);
}